// DenseDiffPooler_81965155877089
// MI455X (gfx1250) — compile-verified
//
#include <hip/hip_runtime.h>

typedef float v2f __attribute__((ext_vector_type(2)));
typedef float v8f __attribute__((ext_vector_type(8)));
typedef int   v4i __attribute__((ext_vector_type(4)));

#define HDIM 256
#define CDIM 512
#define NPG  4096   // nodes per graph

// GEMM tiling
#define BK 32       // K chunk staged per buffer
#define PA 36       // sA row stride (pad): banks (36*r + k) mod 64 distinct for r=0..15
#define PBT 36      // sB (transposed) row stride, same conflict-free property

#if __has_builtin(__builtin_amdgcn_global_load_async_to_lds_b128) && \
    __has_builtin(__builtin_amdgcn_global_load_async_to_lds_b32)  && \
    __has_builtin(__builtin_amdgcn_s_wait_asynccnt)
#define ASYNC_STAGE 1
#else
#define ASYNC_STAGE 0
#endif

// builtin signatures: (T as(1)* src_global, T as(3)* dst_lds, imm offset, imm cpol)
#define AS_GLOBAL_V4(p)  ((__attribute__((address_space(1))) v4i*)(p))
#define AS_LDS_V4(p)     ((__attribute__((address_space(3))) v4i*)(p))
#define AS_GLOBAL_I32(p) ((__attribute__((address_space(1))) int*)(p))
#define AS_LDS_I32(p)    ((__attribute__((address_space(3))) int*)(p))

// ---------------- zero fill ----------------
__global__ void fill_zero(float* __restrict__ p, size_t n) {
  size_t i = (size_t)blockIdx.x * blockDim.x + threadIdx.x;
  size_t stride = (size_t)gridDim.x * blockDim.x;
  for (; i < n; i += stride) p[i] = 0.0f;
}

// ---------------- fp32 WMMA GEMM: Y[N,ncols] = X[N,256] @ W[256,ncols] ----------------
// Block: 256 threads = 8 waves. Block tile 128(M) x 64(N).
// Double-buffered LDS pipeline, async global->LDS staging (b128 for A, scattered b32
// transpose for W so B fragments are single mov-free b64 LDS loads).
// Wave w owns rows [w*16, w*16+16) and all 64 cols (4 v8f accumulators).
__global__ __launch_bounds__(256) void gemm_f32(const float* __restrict__ X,
                                                const float* __restrict__ W,
                                                float* __restrict__ Y,
                                                int ncols) {
  __shared__ float sA[2][128 * PA];  // [row][k]  row-major X tile
  __shared__ float sB[2][64 * PBT];  // [col][k]  transposed W tile

  const int r0   = blockIdx.x * 128;
  const int n0   = blockIdx.y * 64;
  const int tid  = threadIdx.x;
  const int wave = tid >> 5;
  const int lane = tid & 31;
  const int hf   = lane >> 4;   // half-wave: K pair select (0 -> k,k+1 ; 1 -> k+2,k+3)
  const int lm   = lane & 15;

  // per-thread staging coords (fixed across chunks)
  // A tile 128x32: 4 x float4/thread ; W tile 32x64: 8 x float/thread (transposed store)
  int aRow[4], aKc[4];
#pragma unroll
  for (int it = 0; it < 4; ++it) {
    int flat = (it * 256 + tid) * 4;
    aRow[it] = flat >> 5;
    aKc[it]  = flat & 31;
  }
  int bK[8], bC[8];
#pragma unroll
  for (int it = 0; it < 8; ++it) {
    int flat = it * 256 + tid;
    bK[it] = flat >> 6;     // 0..31
    bC[it] = flat & 63;     // 0..63  (consecutive lanes -> consecutive cols: coalesced)
  }

  auto stage = [&](int k0, int buf) {
#if ASYNC_STAGE
#pragma unroll
    for (int it = 0; it < 4; ++it)
      __builtin_amdgcn_global_load_async_to_lds_b128(
          AS_GLOBAL_V4(X + (size_t)(r0 + aRow[it]) * 256 + k0 + aKc[it]),
          AS_LDS_V4(&sA[buf][aRow[it] * PA + aKc[it]]), 0, 0);
#pragma unroll
    for (int it = 0; it < 8; ++it)
      __builtin_amdgcn_global_load_async_to_lds_b32(
          AS_GLOBAL_I32(W + (size_t)(k0 + bK[it]) * ncols + n0 + bC[it]),
          AS_LDS_I32(&sB[buf][bC[it] * PBT + bK[it]]), 0, 0);
#else
#pragma unroll
    for (int it = 0; it < 4; ++it)
      *(float4*)&sA[buf][aRow[it] * PA + aKc[it]] =
          *(const float4*)(X + (size_t)(r0 + aRow[it]) * 256 + k0 + aKc[it]);
#pragma unroll
    for (int it = 0; it < 8; ++it)
      sB[buf][bC[it] * PBT + bK[it]] = W[(size_t)(k0 + bK[it]) * ncols + n0 + bC[it]];
#endif
  };

  v8f acc0 = {}, acc1 = {}, acc2 = {}, acc3 = {};

  stage(0, 0);
  for (int c = 0; c < 256 / BK; ++c) {
    int buf = c & 1;
    if (c < 256 / BK - 1) {
      stage((c + 1) * BK, buf ^ 1);
#if ASYNC_STAGE
      __builtin_amdgcn_s_wait_asynccnt(12);  // 12 newest (next chunk) may remain in flight
#endif
    } else {
#if ASYNC_STAGE
      __builtin_amdgcn_s_wait_asynccnt(0);
#endif
    }
    __syncthreads();

#pragma unroll
    for (int kk = 0; kk < BK; kk += 4) {
      // A frag (16x4 f32): lane l<16 -> A[m][k..k+1], l>=16 -> A[m][k+2..k+3]
      v2f a  = *(const v2f*)&sA[buf][(wave * 16 + lm) * PA + kk + hf * 2];
      // B frag (4x16 f32): lane holds {B[k(+2)][n], B[k+1(+3)][n]} = contiguous in sB[col][k]
      v2f b0 = *(const v2f*)&sB[buf][( 0 + lm) * PBT + kk + hf * 2];
      v2f b1 = *(const v2f*)&sB[buf][(16 + lm) * PBT + kk + hf * 2];
      v2f b2 = *(const v2f*)&sB[buf][(32 + lm) * PBT + kk + hf * 2];
      v2f b3 = *(const v2f*)&sB[buf][(48 + lm) * PBT + kk + hf * 2];
      acc0 = __builtin_amdgcn_wmma_f32_16x16x4_f32(false, a, false, b0, (short)0, acc0, false, false);
      acc1 = __builtin_amdgcn_wmma_f32_16x16x4_f32(false, a, false, b1, (short)0, acc1, false, false);
      acc2 = __builtin_amdgcn_wmma_f32_16x16x4_f32(false, a, false, b2, (short)0, acc2, false, false);
      acc3 = __builtin_amdgcn_wmma_f32_16x16x4_f32(false, a, false, b3, (short)0, acc3, false, false);
    }
    __syncthreads();
  }

  // C/D layout: VGPR v, lanes 0-15 -> row v, lanes 16-31 -> row v+8; col = lane&15
#pragma unroll
  for (int v = 0; v < 8; ++v) {
    int row = r0 + wave * 16 + v + hf * 8;
    float* yp = Y + (size_t)row * ncols + n0 + lm;
    yp[0]  = acc0[v];
    yp[16] = acc1[v];
    yp[32] = acc2[v];
    yp[48] = acc3[v];
  }
}

// ---------------- edge scatter-add: out[dst] += feat[src] ----------------
// One edge per wave; 32-lane coalesced gather + global_atomic_add_f32 scatter.
__global__ __launch_bounds__(256) void edge_aggregate(const float* __restrict__ feat,
                                                      const int* __restrict__ srcI,
                                                      const int* __restrict__ dstI,
                                                      float* __restrict__ out,
                                                      int nE, int nc) {
  int e = blockIdx.x * 8 + (threadIdx.x >> 5);
  if (e >= nE) return;
  int lane = threadIdx.x & 31;
  int s = srcI[e];
  int d = dstI[e];
  const float* sp = feat + (size_t)s * nc;
  float* dp = out + (size_t)d * nc;
  for (int c = lane; c < nc; c += 32) atomicAdd(dp + c, sp[c]);
}

// ---------------- x = relu(x + bias)  (for activations feeding a GEMM) ----------------
__global__ void bias_relu(float* __restrict__ x, const float* __restrict__ b,
                          int nc, size_t total) {
  size_t i = (size_t)blockIdx.x * blockDim.x + threadIdx.x;
  size_t stride = (size_t)gridDim.x * blockDim.x;
  for (; i < total; i += stride) {
    float v = x[i] + b[i & (size_t)(nc - 1)];   // nc is a power of two
    x[i] = v > 0.0f ? v : 0.0f;
  }
}

// ---------------- fused relu(agg+bias) -> softmax row-sum over 512 cols ----------------
// Wave per row, row fully register-resident, wave32 shfl_xor reductions.
__global__ __launch_bounds__(256) void softmax_rowsum(const float* __restrict__ s1raw,
                                                      const float* __restrict__ bias,
                                                      float* __restrict__ t) {
  int row  = blockIdx.x * 8 + (threadIdx.x >> 5);
  int lane = threadIdx.x & 31;
  const float* p = s1raw + (size_t)row * CDIM;
  float v[16];
  float m = -3.402823466e38f;
#pragma unroll
  for (int i = 0; i < 16; ++i) {
    int col = lane + i * 32;
    float x = p[col] + bias[col];
    v[i] = x > 0.0f ? x : 0.0f;
    m = fmaxf(m, v[i]);
  }
#pragma unroll
  for (int off = 16; off; off >>= 1) m = fmaxf(m, __shfl_xor(m, off, 32));
  float D = 0.0f;
#pragma unroll
  for (int i = 0; i < 16; ++i) { v[i] = __expf(v[i] - m); D += v[i]; }
#pragma unroll
  for (int off = 16; off; off >>= 1) D += __shfl_xor(D, off, 32);
  float s = 0.0f;
#pragma unroll
  for (int i = 0; i < 16; ++i) s += v[i] / D;
#pragma unroll
  for (int off = 16; off; off >>= 1) s += __shfl_xor(s, off, 32);
  if (lane == 0) t[row] = s;
}

// ------- fused x1 = relu(agg+bias); out[b,f] += (1/C) * sum_n rowsum[n] * x1[n,f] -------
__global__ __launch_bounds__(256) void final_reduce(const float* __restrict__ x1raw,
                                                    const float* __restrict__ bias,
                                                    const float* __restrict__ t,
                                                    float* __restrict__ out) {
  int b = blockIdx.x;
  int base = b * NPG + blockIdx.y * 256;
  int f = threadIdx.x;                       // 0..255 feature
  float bf = bias[f];
  float acc = 0.0f;
  for (int r = 0; r < 256; ++r) {
    int n = base + r;
    float x = x1raw[(size_t)n * HDIM + f] + bf;
    x = x > 0.0f ? x : 0.0f;
    acc += t[n] * x;
  }
  atomicAdd(out + b * HDIM + f, acc * (1.0f / (float)CDIM));
}

extern "C" void kernel_launch(void* const* d_in, const int* in_sizes, int n_in,
                              void* d_out, int out_size, void* d_ws, size_t ws_size,
                              hipStream_t stream) {
  const float* X   = (const float*)d_in[0];
  const int* eidx  = (const int*)d_in[1];
  const float* Wp1 = (const float*)d_in[2];
  const float* bp1 = (const float*)d_in[3];
  const float* Wp2 = (const float*)d_in[4];
  const float* bp2 = (const float*)d_in[5];
  const float* We1 = (const float*)d_in[6];
  const float* be1 = (const float*)d_in[7];
  const float* We2 = (const float*)d_in[8];
  const float* be2 = (const float*)d_in[9];
  float* out = (float*)d_out;

  const int N  = in_sizes[0] / HDIM;   // 32768
  const int nE = in_sizes[1] / 2;      // 524288
  const int* srcI = eidx;
  const int* dstI = eidx + nE;
  const int nGraphs = N / NPG;         // 8

  float* A  = (float*)d_ws;                 // [N,512] GEMM outputs
  float* Bb = A  + (size_t)N * CDIM;        // [N,512] aggregation outputs
  float* T  = Bb + (size_t)N * CDIM;        // [N] softmax row sums

  dim3 blk(256);
  dim3 gH(N / 128, HDIM / 64);
  dim3 gC(N / 128, CDIM / 64);
  int aggBlocks = (nE + 7) / 8;

  // ---- pool branch: s1 = relu(agg(relu(agg(X@Wp1)+bp1) @ Wp2) + bp2) ----
  gemm_f32<<<gH, blk, 0, stream>>>(X, Wp1, A, HDIM);
  fill_zero<<<2048, 256, 0, stream>>>(Bb, (size_t)N * HDIM);
  edge_aggregate<<<aggBlocks, blk, 0, stream>>>(A, srcI, dstI, Bb, nE, HDIM);
  bias_relu<<<2048, 256, 0, stream>>>(Bb, bp1, HDIM, (size_t)N * HDIM);
  gemm_f32<<<gC, blk, 0, stream>>>(Bb, Wp2, A, CDIM);
  fill_zero<<<2048, 256, 0, stream>>>(Bb, (size_t)N * CDIM);
  edge_aggregate<<<aggBlocks, blk, 0, stream>>>(A, srcI, dstI, Bb, nE, CDIM);
  softmax_rowsum<<<N / 8, blk, 0, stream>>>(Bb, bp2, T);   // bias+relu fused

  // ---- embed branch: x1 = relu(agg(relu(agg(X@We1)+be1) @ We2) + be2) ----
  gemm_f32<<<gH, blk, 0, stream>>>(X, We1, A, HDIM);
  fill_zero<<<2048, 256, 0, stream>>>(Bb, (size_t)N * HDIM);
  edge_aggregate<<<aggBlocks, blk, 0, stream>>>(A, srcI, dstI, Bb, nE, HDIM);
  bias_relu<<<2048, 256, 0, stream>>>(Bb, be1, HDIM, (size_t)N * HDIM);
  gemm_f32<<<gH, blk, 0, stream>>>(Bb, We2, A, HDIM);   // reads Bb, writes A
  fill_zero<<<2048, 256, 0, stream>>>(Bb, (size_t)N * HDIM);
  edge_aggregate<<<aggBlocks, blk, 0, stream>>>(A, srcI, dstI, Bb, nE, HDIM);

  // ---- readout (bias+relu on x1 fused into the reduction) ----
  fill_zero<<<8, 256, 0, stream>>>(out, (size_t)out_size);
  final_reduce<<<dim3(nGraphs, NPG / 256), blk, 0, stream>>>(Bb, be2, T, out);
}